// SubExtractor_29214367548013
// MI455X (gfx1250) — compile-verified
//
#include <hip/hip_runtime.h>
#include <hip/hip_bf16.h>

// ---------------------------------------------------------------------------
// Cluster-attention pooling, restructured for MI455X roofline:
//   S = x @ (Q WK)^T / 16 (+ Qb k-bias term)      4.3 GFLOP
//   P = softmax_c(S);  U[g] += P^T x  (atomics)    4.3 GFLOP
//   Z = U @ WV^T + Q + colsum*bv;  out = relu(Z @ WO^T + bo)   ~1.1 GFLOP
// x read from HBM exactly once (134 MB -> ~5.8us at 23.3 TB/s). fp32 WMMA.
// ---------------------------------------------------------------------------

#define D       256
#define NC      64      // clusters
#define NB      64      // graphs
#define ML      4096    // MAX_NODES pad
#define NNODES  131072
#define TILE_N  64

typedef __attribute__((ext_vector_type(2))) float v2f;
typedef __attribute__((ext_vector_type(8))) float v8f;
typedef __attribute__((ext_vector_type(4))) int   v4i;

typedef __attribute__((address_space(1))) v4i*   as1_v4i;
typedef __attribute__((address_space(3))) v4i*   as3_v4i;
typedef __attribute__((address_space(1))) float* as1_f;
typedef __attribute__((address_space(3))) float* as3_f;

#if __has_builtin(__builtin_amdgcn_wmma_f32_16x16x4_f32)
#define HAVE_WMMA_X4 1
#endif
#if __has_builtin(__builtin_amdgcn_global_load_async_to_lds_b128)
#define HAVE_ASYNC_LDS 1
#endif

// D[16x16] = A[16x4] * B[4x16] + C, wave32.
static __device__ __forceinline__ v8f wmma4(v2f a, v2f b, v8f c) {
#ifdef HAVE_WMMA_X4
  return __builtin_amdgcn_wmma_f32_16x16x4_f32(
      /*neg_a=*/false, a, /*neg_b=*/false, b,
      /*c_mod=*/(short)0, c, /*reuse_a=*/false, /*reuse_b=*/false);
#else
  // shuffle-based fallback (only used if the builtin is missing)
  int lane = threadIdx.x & 31;
  v8f d = c;
#pragma unroll
  for (int k = 0; k < 4; ++k) {
    float bsrc = (k & 1) ? b.y : b.x;
    float bv = __shfl(bsrc, (lane & 15) | ((k & 2) << 3), 32);
#pragma unroll
    for (int r = 0; r < 8; ++r) {
      int m = r + ((lane >> 4) << 3);
      float asrc = (k & 1) ? a.y : a.x;
      float av = __shfl(asrc, m | ((k & 2) << 3), 32);
      d[r] = fmaf(av, bv, d[r]);
    }
  }
  return d;
#endif
}

// -------------------------- tiny setup kernels -----------------------------

__global__ void starts_kernel(const long long* __restrict__ batch,
                              int* __restrict__ starts) {
  int g = threadIdx.x;
  if (g >= NB) return;
  int lo = 0, hi = NNODES;          // lower_bound over sorted batch
  while (lo < hi) {
    int mid = (lo + hi) >> 1;
    if ((int)batch[mid] < g) lo = mid + 1; else hi = mid;
  }
  starts[g] = lo;
}

__global__ void init_kernel(float* __restrict__ U, float* __restrict__ colsum,
                            float* __restrict__ amax, float* __restrict__ mask) {
  int i = blockIdx.x * blockDim.x + threadIdx.x;
  if (i < NB * NC * D) U[i] = 0.0f;
  if (i < NB * NC) colsum[i] = 0.0f;
  if (i < NB * ML) { amax[i] = 0.0f; mask[i] = 0.0f; }
}

// Q[c][i] = Qp[c] . WQ_w[i] + WQ_b[i]
__global__ void q_kernel(const float* __restrict__ Qp, const float* __restrict__ WQ,
                         const float* __restrict__ bq, float* __restrict__ Q) {
  int c = blockIdx.x, i = threadIdx.x;
  const float* qp = Qp + c * D;
  const float* w  = WQ + i * D;
  float s = bq[i];
  for (int j = 0; j < D; ++j) s = fmaf(qp[j], w[j], s);
  Q[c * D + i] = s;
}

// QKs[c][j] = (Q[c] . WK[:,j]) / 16 ; qbs[c] = (Q[c] . bk) / 16
__global__ void qk_kernel(const float* __restrict__ Q, const float* __restrict__ WK,
                          const float* __restrict__ bk, float* __restrict__ QKs,
                          float* __restrict__ qbs) {
  __shared__ float red[256];
  int c = blockIdx.x, j = threadIdx.x;
  const float* q = Q + c * D;
  float s = 0.0f;
  for (int i = 0; i < D; ++i) s = fmaf(q[i], WK[i * D + j], s);
  QKs[c * D + j] = s * 0.0625f;
  red[j] = q[j] * bk[j];
  __syncthreads();
  for (int off = 128; off; off >>= 1) {
    if (j < off) red[j] += red[j + off];
    __syncthreads();
  }
  if (j == 0) qbs[c] = red[0] * 0.0625f;
}

// ------------------------------ main kernel --------------------------------
// Per WG: 64-node tile. S = x QKs^T (+qbs), softmax over clusters, then
// per-graph masked U += P^T x via WMMA + fp32 atomics.

__global__ __launch_bounds__(256) void fused_score_pool(
    const float* __restrict__ x, const long long* __restrict__ batch,
    const int* __restrict__ starts, const float* __restrict__ QKs,
    const float* __restrict__ qbs, float* __restrict__ U,
    float* __restrict__ colsum, float* __restrict__ amax_out,
    float* __restrict__ mask_out) {
  __shared__ __align__(16) float xs[TILE_N * D];   // 64 KB
  __shared__ float P[TILE_N][NC];                  // softmaxed scores, node-major
  __shared__ float Pt[NC][TILE_N + 4];             // masked transpose per graph
  __shared__ int nb[TILE_N];
  __shared__ int npos[TILE_N];

  const int tid  = threadIdx.x;
  const int lane = tid & 31;
  const int wave = tid >> 5;
  const int n0   = blockIdx.x * TILE_N;

  // ---- stage x tile into LDS (async-to-LDS if available) ----
#ifdef HAVE_ASYNC_LDS
  {
    float* gsrc = const_cast<float*>(x) + (size_t)n0 * D;
    for (int i = tid; i < TILE_N * D / 4; i += 256) {
      as1_f gp = (as1_f)(gsrc + i * 4);        // addrspacecast generic->AS1
      as3_f lp = (as3_f)(xs + i * 4);          // addrspacecast generic->AS3
      __builtin_amdgcn_global_load_async_to_lds_b128(
          (as1_v4i)gp, (as3_v4i)lp, 0, 0);
    }
#if __has_builtin(__builtin_amdgcn_s_wait_asynccnt)
    __builtin_amdgcn_s_wait_asynccnt(0);
#else
    asm volatile("s_wait_asynccnt 0" ::: "memory");
#endif
  }
#else
  {
    const float4* gsrc = (const float4*)(x + (size_t)n0 * D);
    for (int i = tid; i < TILE_N * D / 4; i += 256) ((float4*)xs)[i] = gsrc[i];
  }
#endif
  if (tid < TILE_N) {
    int b = (int)batch[n0 + tid];
    nb[tid] = b;
    npos[tid] = n0 + tid - starts[b];
  }
  __syncthreads();

  // ---- S = x_tile @ QKs^T : 16 tiles of 16x16 (4 node-tiles x 4 cluster-tiles)
  for (int t = wave; t < 16; t += 8) {
    int rt = t >> 2, ct = t & 3;
    v8f acc = {};
    const int m = lane & 15, kh = (lane >> 4) << 1;
    const float* arow = &xs[(rt * 16 + m) * D];
    const float* brow = &QKs[(size_t)(ct * 16 + (lane & 15)) * D];
#pragma unroll 4
    for (int k = 0; k < D; k += 4) {
      v2f a; a.x = arow[k + kh]; a.y = arow[k + kh + 1];
      v2f b; b.x = brow[k + kh]; b.y = brow[k + kh + 1];
      acc = wmma4(a, b, acc);
    }
    int col = lane & 15;
#pragma unroll
    for (int r = 0; r < 8; ++r) {
      int row = rt * 16 + r + ((lane < 16) ? 0 : 8);
      int cc  = ct * 16 + col;
      P[row][cc] = acc[r] + qbs[cc];
    }
  }
  __syncthreads();

  // ---- softmax over cluster axis per node; emit argmax + mask ----
  if (tid < TILE_N) {
    float mx = -3.4e38f; int am = 0;
    for (int c = 0; c < NC; ++c) {
      float v = P[tid][c];
      if (v > mx) { mx = v; am = c; }
    }
    float s = 0.0f;
    for (int c = 0; c < NC; ++c) {
      float e = __expf(P[tid][c] - mx);
      P[tid][c] = e; s += e;
    }
    float inv = 1.0f / s;
    for (int c = 0; c < NC; ++c) P[tid][c] *= inv;
    int b = nb[tid], pos = npos[tid];
    if (pos >= 0 && pos < ML) {       // mode='drop'
      amax_out[(size_t)b * ML + pos] = (float)am;
      mask_out[(size_t)b * ML + pos] = 1.0f;
    }
  }
  __syncthreads();

  // ---- per-graph passes (consecutive sorted nodes: usually 1 graph/tile) ----
  const int g0 = nb[0], g1 = nb[TILE_N - 1];
  for (int g = g0; g <= g1; ++g) {
    for (int i = tid; i < NC * TILE_N; i += 256) {
      int n = i >> 6, c = i & 63;
      bool ok = (nb[n] == g) && (npos[n] >= 0) && (npos[n] < ML);
      Pt[c][n] = ok ? P[n][c] : 0.0f;
    }
    __syncthreads();
    // U[g] += Pt @ x_tile : tiles [4 cluster-rowtiles][16 d-coltiles], K=64
    for (int t = wave; t < 64; t += 8) {
      int rt = t >> 4, dt = t & 15;
      v8f acc = {};
      const int m = lane & 15, kh = (lane >> 4) << 1, nn = lane & 15;
      const float* arow = &Pt[rt * 16 + m][0];
#pragma unroll 4
      for (int k = 0; k < TILE_N; k += 4) {
        v2f a; a.x = arow[k + kh]; a.y = arow[k + kh + 1];
        v2f b; b.x = xs[(k + kh) * D + dt * 16 + nn];
        /*   */ b.y = xs[(k + kh + 1) * D + dt * 16 + nn];
        acc = wmma4(a, b, acc);
      }
      int col = lane & 15;
#pragma unroll
      for (int r = 0; r < 8; ++r) {
        int crow = rt * 16 + r + ((lane < 16) ? 0 : 8);
        atomicAdd(&U[((size_t)g * NC + crow) * D + dt * 16 + col], acc[r]);
      }
    }
    if (tid < NC) {
      float s = 0.0f;
      for (int n = 0; n < TILE_N; ++n) s += Pt[tid][n];
      atomicAdd(&colsum[g * NC + tid], s);
    }
    __syncthreads();
  }
}

// ------------------- epilogue GEMMs: [4096,256] @ W^T ----------------------
// mode 0: Out = A@W^T + Q[row%64] + colsum[row]*bias      (Z = U WV^T + ...)
// mode 1: Out = relu(A@W^T + bias)                        (final output)

__global__ __launch_bounds__(256) void gemm_nt_epilogue(
    const float* __restrict__ A, const float* __restrict__ W,
    const float* __restrict__ bias, const float* __restrict__ Qmat,
    const float* __restrict__ colsum, float* __restrict__ Out, int mode) {
  const int lane = threadIdx.x & 31;
  const int wave = threadIdx.x >> 5;
  const int tile = blockIdx.x * 8 + wave;   // 4096 tiles: 256 rowtiles x 16 coltiles
  const int rt = tile >> 4, ct = tile & 15;

  v8f acc = {};
  const int m = lane & 15, kh = (lane >> 4) << 1;
  const float* arow = A + (size_t)(rt * 16 + m) * D;
  const float* brow = W + (size_t)(ct * 16 + (lane & 15)) * D;
#pragma unroll 4
  for (int k = 0; k < D; k += 4) {
    v2f a; a.x = arow[k + kh]; a.y = arow[k + kh + 1];
    v2f b; b.x = brow[k + kh]; b.y = brow[k + kh + 1];
    acc = wmma4(a, b, acc);
  }
  int col = ct * 16 + (lane & 15);
#pragma unroll
  for (int r = 0; r < 8; ++r) {
    int row = rt * 16 + r + ((lane < 16) ? 0 : 8);
    float v = acc[r];
    if (mode == 0) {
      v += Qmat[(size_t)(row & (NC - 1)) * D + col] + colsum[row] * bias[col];
    } else {
      v += bias[col];
      v = fmaxf(v, 0.0f);
    }
    Out[(size_t)row * D + col] = v;
  }
}

// ------------------------------- launcher ----------------------------------

extern "C" void kernel_launch(void* const* d_in, const int* in_sizes, int n_in,
                              void* d_out, int out_size, void* d_ws, size_t ws_size,
                              hipStream_t stream) {
  const float*     x     = (const float*)d_in[0];
  const long long* batch = (const long long*)d_in[1];
  const float*     Qp    = (const float*)d_in[2];
  const float*     WQ_w  = (const float*)d_in[3];
  const float*     WQ_b  = (const float*)d_in[4];
  const float*     WK_w  = (const float*)d_in[5];
  const float*     WK_b  = (const float*)d_in[6];
  const float*     WV_w  = (const float*)d_in[7];
  const float*     WV_b  = (const float*)d_in[8];
  const float*     WO_w  = (const float*)d_in[9];
  const float*     WO_b  = (const float*)d_in[10];

  // workspace layout
  float* Q      = (float*)d_ws;              // 64*256
  float* QKs    = Q + NC * D;                // 64*256
  float* qbs    = QKs + NC * D;              // 64
  int*   starts = (int*)(qbs + NC);          // 64
  float* U      = (float*)(starts + NB);     // 64*64*256
  float* colsum = U + NB * NC * D;           // 64*64
  float* Z      = colsum + NB * NC;          // 4096*256

  // output layout (concatenated flat): out [B,C,D], argmax [B,L], mask [B,L]
  float* out_p  = (float*)d_out;
  float* amax_p = out_p + NB * NC * D;
  float* mask_p = amax_p + NB * ML;

  init_kernel<<<(NB * NC * D + 255) / 256, 256, 0, stream>>>(U, colsum, amax_p, mask_p);
  starts_kernel<<<1, 64, 0, stream>>>(batch, starts);
  q_kernel<<<NC, D, 0, stream>>>(Qp, WQ_w, WQ_b, Q);
  qk_kernel<<<NC, D, 0, stream>>>(Q, WK_w, WK_b, QKs, qbs);

  fused_score_pool<<<NNODES / TILE_N, 256, 0, stream>>>(
      x, batch, starts, QKs, qbs, U, colsum, amax_p, mask_p);

  gemm_nt_epilogue<<<512, 256, 0, stream>>>(U, WV_w, WV_b, Q, colsum, Z, 0);
  gemm_nt_epilogue<<<512, 256, 0, stream>>>(Z, WO_w, WO_b, Q, colsum, out_p, 1);
}